// VanillaGNNLayer_34772055228944
// MI455X (gfx1250) — compile-verified
//
#include <hip/hip_runtime.h>

typedef __attribute__((ext_vector_type(2))) float v2f;
typedef __attribute__((ext_vector_type(8))) float v8f;

#define D 128  // DIN == DOUT == 128

// ---------------------------------------------------------------------------
// Kernel 1: h = x @ W^T via V_WMMA_F32_16X16X4_F32 (full fp32, exact).
// Block: 256 threads = 8 waves. Block owns 16 rows of x (contiguous 8KB ->
// LDS). Wave w owns output-feature tile [w*16, w*16+16). K=128 in steps of 4.
// Also seeds out = h (the identity/self term of the GNN aggregation).
// ---------------------------------------------------------------------------
__global__ __launch_bounds__(256) void gnn_gemm_wmma(
    const float* __restrict__ x, const float* __restrict__ W,
    float* __restrict__ h, float* __restrict__ out, int N)
{
  __shared__ float xs[16 * D];

  const int row0 = blockIdx.x * 16;

  // 16 full rows of x are contiguous: 2048 floats = 512 float4; 256 threads.
  const float4* src = (const float4*)(x + (size_t)row0 * D);
  float4* dst = (float4*)xs;
  dst[threadIdx.x]       = src[threadIdx.x];
  dst[threadIdx.x + 256] = src[threadIdx.x + 256];
  __syncthreads();

  const int wave = threadIdx.x >> 5;   // 0..7 -> output col tile
  const int lane = threadIdx.x & 31;
  const int half = lane >> 4;          // lane half selects K sub-pair / M+8
  const int l    = lane & 15;
  const int j0   = wave * 16;

  // A 16x4 f32: lanes 0-15 -> M=l, VGPR{0,1} = K{k0,k0+1};
  //             lanes 16-31 -> VGPR{0,1} = K{k0+2,k0+3}.
  // B 4x16 f32: B[k][n] = W[j0+n][k]; same lane-half K split as A.
  const float* xrow = xs + l * D;
  const float* wrow = W + (size_t)(j0 + l) * D;

  v8f c = {};
  #pragma unroll
  for (int k0 = 0; k0 < D; k0 += 4) {
    const int ka = k0 + half * 2;
    v2f a = { xrow[ka], xrow[ka + 1] };
    v2f b = { wrow[ka], wrow[ka + 1] };
    // 8 args: (neg_a, A, neg_b, B, c_mod, C, reuse_a, reuse_b)
    c = __builtin_amdgcn_wmma_f32_16x16x4_f32(
        false, a, false, b, (short)0, c, false, false);
  }

  // C/D 16x16 f32 layout: VGPR r -> M = r (lanes 0-15) / r+8 (lanes 16-31).
  const int mbase = row0 + half * 8;
  #pragma unroll
  for (int r = 0; r < 8; ++r) {
    const int row = mbase + r;
    if (row < N) {
      const size_t o = (size_t)row * D + j0 + l;
      const float v = c[r];
      h[o]   = v;   // scratch copy read by the scatter kernel
      out[o] = v;   // identity term: out starts as h
    }
  }
}

// ---------------------------------------------------------------------------
// Kernel 2: out[row[e]] += h[col[e]] for all edges (duplicates accumulate,
// matching segment_sum). One wave per edge: 32 lanes x float4 = coalesced
// 512B gather; 4 fp32 L2 atomics per lane. h and out (51MB each) are
// L2-resident on MI455X (192MB), so the random traffic never hits HBM.
// ---------------------------------------------------------------------------
__global__ __launch_bounds__(256) void gnn_edge_scatter(
    const int* __restrict__ rows, const int* __restrict__ cols,
    const float* __restrict__ h, float* __restrict__ out, long long E)
{
  const long long tid = (long long)blockIdx.x * 256 + threadIdx.x;
  const long long e = tid >> 5;            // same edge for all 32 lanes
  if (e >= E) return;
  const int chunk = (int)(tid & 31);       // lane -> 4-float chunk of the row

  const int r = rows[e];                   // broadcast load (same addr/wave)
  const int s = cols[e];

  const float4 v = *(const float4*)(h + (size_t)s * D + chunk * 4);
  float* p = out + (size_t)r * D + chunk * 4;
  atomicAdd(p + 0, v.x);                   // global_atomic_add_f32, no return
  atomicAdd(p + 1, v.y);
  atomicAdd(p + 2, v.z);
  atomicAdd(p + 3, v.w);
}

extern "C" void kernel_launch(void* const* d_in, const int* in_sizes, int n_in,
                              void* d_out, int out_size, void* d_ws, size_t ws_size,
                              hipStream_t stream) {
  const float* x  = (const float*)d_in[0];
  const int*   ei = (const int*)d_in[1];   // [2, E] flat: rows then cols
  const float* W  = (const float*)d_in[2];
  float* out = (float*)d_out;
  float* h   = (float*)d_ws;               // N*D*4 = 51.2 MB scratch

  const int       N = in_sizes[0] / D;     // 100000
  const long long E = (long long)in_sizes[1] / 2;

  const int gemm_blocks = (N + 15) / 16;
  gnn_gemm_wmma<<<gemm_blocks, 256, 0, stream>>>(x, W, h, out, N);

  const long long work = E * 32;
  const int eblocks = (int)((work + 255) / 256);
  gnn_edge_scatter<<<eblocks, 256, 0, stream>>>(ei, ei + E, h, out, E);
}